// CRF_41927470743974
// MI455X (gfx1250) — compile-verified
//
#include <hip/hip_runtime.h>

// CRF NLL forward pass on gfx1250.
// Step recursion recast as fp32 GEMM:  s[b,j] = sum_i exp(alpha[b,i]-m[b]) * exp(T[i,j])
// -> V_WMMA_F32_16X16X4_F32, B-fragments (exp(T)) held in registers across all steps.

typedef __attribute__((ext_vector_type(2))) float v2f;
typedef __attribute__((ext_vector_type(8))) float v8f;

#define B_   128
#define S_   1024
#define L_   128
#define MT   16     // batch tile per workgroup (WMMA M)
#define PSTR 132    // padded LDS row stride in floats (132*4 bytes: 8B aligned, bank-shift 4)

__global__ void crf_init_out(float* out) { out[0] = 0.0f; }

// ---------------------------------------------------------------------------
// Gold path score (numerator) per batch: one wave per batch.
// ---------------------------------------------------------------------------
__global__ __launch_bounds__(256) void crf_numerator(
    const float* __restrict__ logits, const float* __restrict__ trans,
    const int* __restrict__ tags, const int* __restrict__ mask,
    float* __restrict__ numer)
{
    const int wave = threadIdx.x >> 5;
    const int lane = threadIdx.x & 31;
    const int b = blockIdx.x * 8 + wave;
    if (b >= B_) return;

    const float* lg = logits + (size_t)b * S_ * L_;
    const int*   tg = tags + b * S_;
    const int*   mk = mask + b * S_;

    float emit = 0.0f, tr = 0.0f;
    int cnt = 0;
    for (int t = lane; t < S_; t += 32) {
        const int   mt  = mk[t];
        const float mf  = mt ? 1.0f : 0.0f;
        const int   tag = tg[t];
        if (t < S_ - 1) emit += lg[t * L_ + tag] * mf;        // emit sum over t<S-1
        if (t >= 1)     tr   += trans[tg[t - 1] * L_ + tag] * mf; // trans pairs masked by mask[t]
        cnt += mt ? 1 : 0;
    }
    #pragma unroll
    for (int m = 16; m >= 1; m >>= 1) {
        emit += __shfl_xor(emit, m, 32);
        tr   += __shfl_xor(tr,   m, 32);
        cnt  += __shfl_xor(cnt,  m, 32);
    }
    if (lane == 0) {
        const int   last_idx = cnt - 1;
        const int   lt       = tg[last_idx < 0 ? 0 : last_idx];
        const float mlast    = mk[S_ - 1] ? 1.0f : 0.0f;
        numer[b] = emit + tr + lg[(S_ - 1) * L_ + lt] * mlast;
    }
}

// ---------------------------------------------------------------------------
// Forward (alpha) recursion. One workgroup = 16 batches, 8 waves; wave w owns
// output label columns [16w, 16w+16). E = exp(T) tiles live in VGPRs.
// ---------------------------------------------------------------------------
__global__ __launch_bounds__(256) void crf_forward(
    const float* __restrict__ logits, const float* __restrict__ trans,
    const int* __restrict__ mask, const float* __restrict__ numer,
    float* __restrict__ out)
{
    __shared__ float aL[MT * PSTR];   // alpha[16][L] padded
    __shared__ float pL[MT * PSTR];   // p = exp(alpha - m)
    __shared__ float mL[MT];          // per-batch running max

    const int tid   = threadIdx.x;
    const int wave  = tid >> 5;
    const int lane  = tid & 31;
    const int bBase = blockIdx.x * MT;

    const int ncol  = wave * 16 + (lane & 15);  // global label column for this lane's B/D tile
    const int khalf = (lane >> 4) << 1;         // 0 (lanes 0-15) or 2 (lanes 16-31)
    const int arow  = lane & 15;                // A-fragment batch row
    const int mbase = (lane >> 4) << 3;         // D rows: r + 0 or r + 8

    // --- Preload loop-invariant B fragments: E[i][ncol] = exp(T[i][ncol]) ---
    // 16x4 fp32 B layout: vgpr0 = rows {4k+0 | lanes<16 ; 4k+2 | lanes>=16}, vgpr1 = +1.
    v2f Bf[32];
    #pragma unroll
    for (int k = 0; k < 32; ++k) {
        const int r = 4 * k + khalf;
        Bf[k].x = __expf(trans[r * L_ + ncol]);
        Bf[k].y = __expf(trans[(r + 1) * L_ + ncol]);
    }

    // --- alpha0 = logits[:, 0, :] ---
    const int bl = tid >> 4;   // local batch 0..15 (16 threads per batch)
    const int i0 = tid & 15;
    {
        const float* lg0 = logits + (size_t)(bBase + bl) * S_ * L_;
        #pragma unroll
        for (int k = 0; k < 8; ++k)
            aL[bl * PSTR + i0 + 16 * k] = lg0[i0 + 16 * k];
    }
    __syncthreads();

    for (int t = 1; t < S_; ++t) {
        // ---- Phase A: per-batch max, p = exp(alpha - m) ----
        float v[8];
        float mx = -3.402823466e38f;
        #pragma unroll
        for (int k = 0; k < 8; ++k) {
            v[k] = aL[bl * PSTR + i0 + 16 * k];
            mx = fmaxf(mx, v[k]);
        }
        #pragma unroll
        for (int m = 8; m >= 1; m >>= 1)
            mx = fmaxf(mx, __shfl_xor(mx, m, 32));
        #pragma unroll
        for (int k = 0; k < 8; ++k)
            pL[bl * PSTR + i0 + 16 * k] = __expf(v[k] - mx);
        if (i0 == 0) mL[bl] = mx;
        __syncthreads();

        // ---- Prefetch this step's logits for our D tile (and hint next step) ----
        float lg[8];
        #pragma unroll
        for (int r = 0; r < 8; ++r) {
            const int mloc = r + mbase;
            lg[r] = logits[((size_t)(bBase + mloc) * S_ + t) * L_ + ncol];
        }
        {
            const int tn = (t + 1 < S_) ? (t + 1) : t;
            __builtin_prefetch(&logits[((size_t)(bBase + mbase) * S_ + tn) * L_ + ncol], 0, 0);
        }

        // ---- Phase B: D(16x16) = P(16x128) x E(128x16), two independent acc chains ----
        v8f acc0 = {0.f, 0.f, 0.f, 0.f, 0.f, 0.f, 0.f, 0.f};
        v8f acc1 = {0.f, 0.f, 0.f, 0.f, 0.f, 0.f, 0.f, 0.f};
        #pragma unroll
        for (int k = 0; k < 32; k += 2) {
            v2f a0, a1;
            const float* s0 = &pL[arow * PSTR + 4 * k + khalf];
            const float* s1 = &pL[arow * PSTR + 4 * (k + 1) + khalf];
            a0.x = s0[0]; a0.y = s0[1];
            a1.x = s1[0]; a1.y = s1[1];
            acc0 = __builtin_amdgcn_wmma_f32_16x16x4_f32(
                false, a0, false, Bf[k], (short)0, acc0, false, false);
            acc1 = __builtin_amdgcn_wmma_f32_16x16x4_f32(
                false, a1, false, Bf[k + 1], (short)0, acc1, false, false);
        }

        // ---- Phase C: alpha' = logit + m + log(s), honoring mask ----
        #pragma unroll
        for (int r = 0; r < 8; ++r) {
            const int mloc = r + mbase;
            const int mk   = mask[(size_t)(bBase + mloc) * S_ + t];
            if (mk) {
                const float s  = acc0[r] + acc1[r];
                aL[mloc * PSTR + ncol] = lg[r] + mL[mloc] + __logf(s);
            }
        }
        __syncthreads();
    }

    // ---- log_denominator[b] = logsumexp_j alpha[b,j]; accumulate NLL ----
    {
        float v[8];
        float mx = -3.402823466e38f;
        #pragma unroll
        for (int k = 0; k < 8; ++k) {
            v[k] = aL[bl * PSTR + i0 + 16 * k];
            mx = fmaxf(mx, v[k]);
        }
        #pragma unroll
        for (int m = 8; m >= 1; m >>= 1)
            mx = fmaxf(mx, __shfl_xor(mx, m, 32));
        float s = 0.0f;
        #pragma unroll
        for (int k = 0; k < 8; ++k)
            s += __expf(v[k] - mx);
        #pragma unroll
        for (int m = 8; m >= 1; m >>= 1)
            s += __shfl_xor(s, m, 32);
        if (i0 == 0) {
            const float logZ = mx + __logf(s);
            atomicAdd(out, logZ - numer[bBase + bl]);
        }
    }
}

// ---------------------------------------------------------------------------
extern "C" void kernel_launch(void* const* d_in, const int* in_sizes, int n_in,
                              void* d_out, int out_size, void* d_ws, size_t ws_size,
                              hipStream_t stream)
{
    const float* logits = (const float*)d_in[0];   // (B,S,L) fp32
    const float* trans  = (const float*)d_in[1];   // (L,L)   fp32
    const int*   tags   = (const int*)d_in[2];     // (B,S)
    const int*   mask   = (const int*)d_in[3];     // (B,S)
    float*       out    = (float*)d_out;           // scalar
    float*       numer  = (float*)d_ws;            // B floats scratch

    crf_init_out<<<1, 1, 0, stream>>>(out);
    crf_numerator<<<B_ / 8, 256, 0, stream>>>(logits, trans, tags, mask, numer);
    crf_forward<<<B_ / MT, 256, 0, stream>>>(logits, trans, mask, numer, out);
}